// ANFISRuleStrengthLayer_85203561218828
// MI455X (gfx1250) — compile-verified
//
#include <hip/hip_runtime.h>

// ANFIS rule-strength layer == batched 64x64 outer product:
//   out[b, j1*64 + j2] = x[b,0,j1] * x[b,1,j2],  b in [0,32768)
// Store-bandwidth-bound (512 MB out vs 0.13 GFLOP). We express each 16x16
// output tile as a rank-1 WMMA: D = A(16x4) x B(4x16) with only K=0 populated.

typedef __attribute__((ext_vector_type(2))) float v2f;
typedef __attribute__((ext_vector_type(8))) float v8f;

#define NMF    64
#define CONSEQ 4096   // 64*64
#define TILES_PER_B 16  // 4x4 tiles of 16x16

__global__ __launch_bounds__(256)
void anfis_outer_wmma(const float* __restrict__ x,  // [B, 2, 64]
                      float* __restrict__ out,      // [B, 4096]
                      int nTiles) {
  const int waveInBlock = (int)(threadIdx.x >> 5);
  const int lane        = (int)(threadIdx.x & 31);
  const int wave = (int)blockIdx.x * ((int)blockDim.x >> 5) + waveInBlock;
  if (wave >= nTiles) return;  // uniform per wave -> EXEC all-1s where WMMA runs

  const int b    = wave >> 4;        // batch row
  const int tile = wave & 15;        // which 16x16 tile of the 64x64 matrix
  const int tm   = (tile >> 2) * 16; // j1 (row) tile base
  const int tn   = (tile & 3) * 16;  // j2 (col) tile base

  const float* __restrict__ x0 = x + (size_t)b * (2 * NMF);      // x[b,0,:]
  const float* __restrict__ x1 = x0 + NMF;                       // x[b,1,:]

  const int  l16 = lane & 15;
  const bool lo  = lane < 16;

  // A (16x4 f32, 2 VGPRs): VGPR0 lanes0-15 = K=0 column (x0 tile), rest = 0.
  // B (4x16 f32, 2 VGPRs): VGPR0 lanes0-15 = K=0 row (x1 tile); K>=1 rows are
  // don't-care because A is zero there -> set to 0 anyway.
  float av = x0[tm + l16];
  float bv = x1[tn + l16];
  v2f A; A.x = lo ? av : 0.0f; A.y = 0.0f;
  v2f B; B.x = lo ? bv : 0.0f; B.y = 0.0f;
  v8f C = {};

  // D[m][n] = x0[tm+m] * x1[tn+n]
  v8f D = __builtin_amdgcn_wmma_f32_16x16x4_f32(
      /*neg_a=*/false, A, /*neg_b=*/false, B,
      /*c_mod=*/(short)0, C, /*reuse_a=*/false, /*reuse_b=*/false);

  // C/D layout: VGPR r -> row M=r (lanes 0-15) and M=r+8 (lanes 16-31), N=lane&15.
  // Each b32 store = two contiguous 64B segments; output is streamed once and
  // never re-read -> non-temporal stores keep the 192MB L2 for inputs.
  float* dst = out + (size_t)b * CONSEQ
                   + (size_t)(tm + (lane >> 4) * 8) * NMF
                   + (size_t)(tn + l16);
#pragma unroll
  for (int r = 0; r < 8; ++r) {
    __builtin_nontemporal_store(D[r], dst + (size_t)r * NMF);
  }
}

extern "C" void kernel_launch(void* const* d_in, const int* in_sizes, int n_in,
                              void* d_out, int out_size, void* d_ws, size_t ws_size,
                              hipStream_t stream) {
  const float* x = (const float*)d_in[0];
  float* out = (float*)d_out;

  const int batch  = in_sizes[0] / (2 * NMF);   // 32768
  const int nTiles = batch * TILES_PER_B;       // 524288 waves
  const int threads = 256;                      // 8 waves per block
  const int wavesPerBlock = threads / 32;
  const int blocks = (nTiles + wavesPerBlock - 1) / wavesPerBlock;

  anfis_outer_wmma<<<blocks, threads, 0, stream>>>(x, out, nTiles);
}